// PointcloudToVoxels_6347961663764
// MI455X (gfx1250) — compile-verified
//
#include <hip/hip_runtime.h>
#include <math.h>

// ---------------------------------------------------------------------------
// PointcloudToVoxels for MI455X (gfx1250)
//
// Problem constants (from the reference):
//   B=4 batches, N=480*640 points, C=16 channels, grid 96x96x48 (V=442368)
//   voxel = floor((p - origin)/0.05); occupancy = count >= 10
//   data[c][v] = max over points in voxel v of attrs[c], else 0
//
// Strategy (scatter/atomic workload — WMMA does not apply):
//   1) init:      scatter target = -inf (max identity), counts = 0
//   2) scatter:   per point: 1x global_atomic_add_f32 (count)
//                          16x global_atomic_max_num_f32 (channel maxima)
//                 both at DEV scope (cross-WGP coherent, resolved in L2).
//                 Inputs streamed with non-temporal loads so the ~120MB
//                 atomic target stays resident in the 192MB L2.
//   3) finalize:  occ = cnt>=10; data = occ && finite ? vmax : 0
//
// If the workspace is big enough we scatter into a [B][V][C] layout so each
// point's 16 atomics land in ONE 64-byte line, then transpose to the output
// [B][C][V] layout in the finalize pass (coalesced both ways). Otherwise we
// scatter straight into d_out (identical math, more L2 line traffic).
// ---------------------------------------------------------------------------

#define GRID_W 96
#define GRID_L 96
#define GRID_H 48
#define NVOX   (GRID_W * GRID_L * GRID_H)   // 442368
#define BATCH  4
#define CH     16
#define NPTS   (480 * 640)                  // 307200
#define VOXSZ  0.05f
#define MINPTS 10.0f

// ---- native CDNA5 non-returning FP32 atomics, device scope -----------------
// (non-returning => tracked on STOREcnt, no wait needed inside the kernel;
//  scope:SCOPE_DEV => RMW resolves at the device-coherent L2 atomic units)
__device__ __forceinline__ void gmax_f32(float* addr, float val) {
  asm volatile("global_atomic_max_num_f32 %0, %1, off scope:SCOPE_DEV"
               :: "v"(addr), "v"(val) : "memory");
}
__device__ __forceinline__ void gadd_f32(float* addr, float val) {
  asm volatile("global_atomic_add_f32 %0, %1, off scope:SCOPE_DEV"
               :: "v"(addr), "v"(val) : "memory");
}

// ---- kernel 1: fill first n_max4 float4's with -inf, next n_cnt4 with 0 ----
// (max-buffer and count-buffer are contiguous in both layouts)
__global__ void vox_init_kernel(float4* __restrict__ base) {
  const size_t n_max4 = (size_t)BATCH * CH * NVOX / 4;   // 7,077,888
  const size_t n_tot4 = n_max4 + (size_t)BATCH * NVOX / 4;
  size_t i = (size_t)blockIdx.x * blockDim.x + threadIdx.x;
  if (i >= n_tot4) return;
  const float m = -INFINITY;
  float4 v = (i < n_max4) ? make_float4(m, m, m, m)
                          : make_float4(0.f, 0.f, 0.f, 0.f);
  base[i] = v;
}

// ---- kernel 2: per-point scatter ------------------------------------------
// maxbuf element (b,v,c) lives at b*C*V + v*SV + c*SC
//   workspace layout [B][V][C]: SV=CH, SC=1   (16 atomics -> one 64B line)
//   direct    layout [B][C][V]: SV=1,  SC=V
template <int SV, int SC>
__global__ void vox_scatter_kernel(const float* __restrict__ coords,   // [B][3][N]
                                   const float* __restrict__ attrs,    // [B][C][N]
                                   const float* __restrict__ origin,   // [B][3]
                                   float* __restrict__ maxbuf,
                                   float* __restrict__ cntbuf) {       // [B][V]
  int tid = blockIdx.x * blockDim.x + threadIdx.x;
  if (tid >= BATCH * NPTS) return;
  int b = tid / NPTS;
  int n = tid - b * NPTS;

  const float* cb = coords + (size_t)b * 3 * NPTS + n;
  // stream coords non-temporally: read-once data, keep L2 for the atomics
  float x = __builtin_nontemporal_load(cb);
  float y = __builtin_nontemporal_load(cb + NPTS);
  float z = __builtin_nontemporal_load(cb + 2 * NPTS);

  float ox = origin[b * 3 + 0];
  float oy = origin[b * 3 + 1];
  float oz = origin[b * 3 + 2];

  float fx = floorf((x - ox) / VOXSZ);
  float fy = floorf((y - oy) / VOXSZ);
  float fz = floorf((z - oz) / VOXSZ);

  bool ok = (fx >= 0.f) & (fx < (float)GRID_W) &
            (fy >= 0.f) & (fy < (float)GRID_L) &
            (fz >= 0.f) & (fz < (float)GRID_H);
  if (!ok) return;

  int flat = (int)fx * (GRID_L * GRID_H) + (int)fy * GRID_H + (int)fz;

  gadd_f32(cntbuf + (size_t)b * NVOX + flat, 1.0f);

  const float* ab = attrs + (size_t)b * CH * NPTS + n;
  float* mb = maxbuf + (size_t)b * CH * NVOX + (size_t)flat * SV;
#pragma unroll
  for (int c = 0; c < CH; ++c) {
    float v = __builtin_nontemporal_load(ab + (size_t)c * NPTS);
    gmax_f32(mb + (size_t)c * SC, v);
  }
}

// ---- kernel 3a: finalize from [B][V][C] workspace, transpose to output ----
__global__ void vox_finalize_ws_kernel(const float* __restrict__ ws_max, // [B][V][C]
                                       const float* __restrict__ ws_cnt, // [B][V]
                                       float* __restrict__ out_data,     // [B][C][V]
                                       float* __restrict__ out_occ) {    // [B][V]
  int tid = blockIdx.x * blockDim.x + threadIdx.x;
  if (tid >= BATCH * NVOX) return;
  int b = tid / NVOX;
  int v = tid - b * NVOX;

  float cnt = ws_cnt[tid];
  float occ = (cnt >= MINPTS) ? 1.0f : 0.0f;
  out_occ[tid] = occ;

  // 16 contiguous channel maxima for this voxel: 4x float4 loads
  const float4* src = (const float4*)(ws_max + (size_t)tid * CH);
  float4 q0 = src[0], q1 = src[1], q2 = src[2], q3 = src[3];
  float vals[CH] = {q0.x, q0.y, q0.z, q0.w, q1.x, q1.y, q1.z, q1.w,
                    q2.x, q2.y, q2.z, q2.w, q3.x, q3.y, q3.z, q3.w};

  float* dst = out_data + (size_t)b * CH * NVOX + v;
#pragma unroll
  for (int c = 0; c < CH; ++c) {
    float val = vals[c];
    bool good = (occ != 0.0f) && isfinite(val);
    dst[(size_t)c * NVOX] = good ? val : 0.0f;   // coalesced per channel
  }
}

// ---- kernel 3b: finalize in place on d_out ([B][C][V] was scatter target) --
__global__ void vox_finalize_inplace_kernel(float* __restrict__ out_data,  // [B][C][V]
                                            float* __restrict__ out_occ) { // [B][V]
  int tid = blockIdx.x * blockDim.x + threadIdx.x;
  if (tid >= BATCH * NVOX) return;
  int b = tid / NVOX;
  int v = tid - b * NVOX;

  float cnt = out_occ[tid];
  float occ = (cnt >= MINPTS) ? 1.0f : 0.0f;
  out_occ[tid] = occ;

  float* dst = out_data + (size_t)b * CH * NVOX + v;
#pragma unroll
  for (int c = 0; c < CH; ++c) {
    float val = dst[(size_t)c * NVOX];
    bool good = (occ != 0.0f) && isfinite(val);
    dst[(size_t)c * NVOX] = good ? val : 0.0f;
  }
}

extern "C" void kernel_launch(void* const* d_in, const int* in_sizes, int n_in,
                              void* d_out, int out_size, void* d_ws, size_t ws_size,
                              hipStream_t stream) {
  const float* coords = (const float*)d_in[0];   // [4][3][480][640]
  const float* attrs  = (const float*)d_in[1];   // [4][16][480][640]
  const float* origin = (const float*)d_in[2];   // [4][3]

  float* out      = (float*)d_out;
  float* out_data = out;                                   // B*C*V floats
  float* out_occ  = out + (size_t)BATCH * CH * NVOX;       // B*V floats

  const size_t need_ws =
      ((size_t)BATCH * CH * NVOX + (size_t)BATCH * NVOX) * sizeof(float);
  const bool use_ws = (ws_size >= need_ws);   // fixed per harness -> deterministic

  const int TB = 256;  // 8 wave32 waves
  const int init_blocks     = (int)(((size_t)BATCH * (CH + 1) * NVOX / 4 + TB - 1) / TB);
  const int scatter_blocks  = (BATCH * NPTS + TB - 1) / TB;     // 4800
  const int finalize_blocks = (BATCH * NVOX + TB - 1) / TB;     // 6912

  if (use_ws) {
    float* ws_max = (float*)d_ws;                            // [B][V][C]
    float* ws_cnt = ws_max + (size_t)BATCH * CH * NVOX;      // [B][V]

    vox_init_kernel<<<init_blocks, TB, 0, stream>>>((float4*)d_ws);
    vox_scatter_kernel<CH, 1><<<scatter_blocks, TB, 0, stream>>>(
        coords, attrs, origin, ws_max, ws_cnt);
    vox_finalize_ws_kernel<<<finalize_blocks, TB, 0, stream>>>(
        ws_max, ws_cnt, out_data, out_occ);
  } else {
    vox_init_kernel<<<init_blocks, TB, 0, stream>>>((float4*)d_out);
    vox_scatter_kernel<1, NVOX><<<scatter_blocks, TB, 0, stream>>>(
        coords, attrs, origin, out_data, out_occ);
    vox_finalize_inplace_kernel<<<finalize_blocks, TB, 0, stream>>>(
        out_data, out_occ);
  }
}